// ManifoldConstrainedHyperConnections_66374424592794
// MI455X (gfx1250) — compile-verified
//
#include <hip/hip_runtime.h>
#include <math.h>

typedef __attribute__((ext_vector_type(2))) float v2f;
typedef __attribute__((ext_vector_type(8))) float v8f;

namespace {
constexpr int kS = 4;
constexpr int kV = 1;
constexpr int kB = 2;
constexpr int kN = 4096;
constexpr int kD = 2048;
constexpr int kSD = kS * kD;                 // 8192
constexpr int kAlphaCols = kS * (kS + kV);   // 20
constexpr int kCols = kAlphaCols + kS;       // 24
constexpr int kPack = 32;                    // padded B width (two 16-wide WMMA tiles)
constexpr int kTokens = kB * kN;             // 8192
constexpr int kTile = 16;                    // tokens per block (WMMA M)
constexpr int kChunk = 256;                  // K elements staged per iteration
constexpr int kAStride = kChunk + 1;         // LDS pad
constexpr int kIters = 20;
}

__device__ __forceinline__ float sigmoidf_(float x) { return 1.0f / (1.0f + expf(-x)); }

__device__ __forceinline__ float4 f4_fma(float a, const float4 x, const float4 acc) {
    float4 r;
    r.x = fmaf(a, x.x, acc.x);
    r.y = fmaf(a, x.y, acc.y);
    r.z = fmaf(a, x.z, acc.z);
    r.w = fmaf(a, x.w, acc.w);
    return r;
}

// ---------------------------------------------------------------------------
// Kernel 1: pack dyn_alpha_fn (8192x20) + dyn_beta_fn (8192x4) into an
// 8192x32 matrix stored directly in WMMA B-fragment order:
//   element (k, j) -> bswz[(k>>2)*128 + ((k>>1)&1)*64 + j*2 + (k&1)]
// so a lane's (k, k+1) pair is a contiguous 8-byte v2f, and a 256-K chunk is
// one contiguous 32 KB panel (async-copyable to LDS).
// ---------------------------------------------------------------------------
__global__ __launch_bounds__(256) void mchc_pack(const float* __restrict__ dyn_alpha,
                                                 const float* __restrict__ dyn_beta,
                                                 float* __restrict__ bswz) {
    const int idx = blockIdx.x * 256 + threadIdx.x;
    if (idx >= kSD * kPack) return;
    const int k = idx >> 5;
    const int j = idx & 31;
    float v = 0.0f;
    if (j < kAlphaCols) v = dyn_alpha[(size_t)k * kAlphaCols + j];
    else if (j < kCols) v = dyn_beta[(size_t)k * kS + (j - kAlphaCols)];
    const int q = k >> 2;           // K-quad index
    const int kh = (k >> 1) & 1;    // fragment half (lanes 0-15 vs 16-31)
    const int kv = k & 1;           // fragment vgpr (pair element)
    bswz[(size_t)q * 128 + kh * 64 + j * 2 + kv] = v;
}

// ---------------------------------------------------------------------------
// Kernel 2: per 16-token tile, the 24 dot-products of flat*(gamma+1) against
// the packed matrix via V_WMMA_F32_16X16X4_F32 (+ sum-of-squares for the
// norm), then per-token 4x4 Sinkhorn + sigmoids -> 24 coefficients.
// B panels are staged to LDS with GLOBAL_LOAD_ASYNC_TO_LDS_B128.
// ---------------------------------------------------------------------------
__global__ __launch_bounds__(256) void mchc_gemv_sinkhorn(
    const float* __restrict__ residuals,
    const float* __restrict__ gamma,
    const float* __restrict__ bswz,
    const float* __restrict__ static_alpha,
    const float* __restrict__ pre_branch_scale,
    const float* __restrict__ residual_scale,
    const float* __restrict__ static_beta,
    const float* __restrict__ h_post_scale,
    float* __restrict__ tokrec) {
    __shared__ float a_lds[kTile * kAStride];   // weighted A chunk, 16 x 256 (+pad)
    __shared__ float b_lds[kChunk * kPack];     // swizzled B panel, 32 KB
    __shared__ float c_acc[2 * 16 * 16];        // cross-wave C reduction
    __shared__ float ssq_lds[16 * 16];

    const int tid = threadIdx.x;
    const int tok0 = blockIdx.x * kTile;

    // zero C accumulators (512 floats, 256 threads)
    c_acc[tid] = 0.0f;
    c_acc[tid + 256] = 0.0f;

    // staging role: 16 threads per token, float4 loads
    const int m = tid >> 4;
    const int g = tid & 15;
    const int token = tok0 + m;
    const int b_ = token >> 12;          // token / 4096
    const int n_ = token & (kN - 1);

    // WMMA role
    const int lane = tid & 31;
    const int wave = tid >> 5;
    const int mrow = lane & 15;                   // A fragment row (M)
    const int khalf = (lane >= 16) ? 2 : 0;       // A fragment K half
    const int ncol = lane & 15;                   // B/C fragment col (N)
    // B fragment lane base (floats): wave's q-slice + half + column pair
    const int bbase = wave * 1024 + ((lane >= 16) ? 64 : 0) + ncol * 2;

    // LDS byte offset of the B panel + per-thread async-copy target offsets
    const unsigned b_lds_base = (unsigned)(size_t)(&b_lds[0]);

    float ssq = 0.0f;
    v8f c0 = {};
    v8f c1 = {};

    for (int chunk = 0; chunk < kSD / kChunk; ++chunk) {
        __syncthreads();  // previous chunk's fragment reads done / init visible
        const int i0 = chunk * kChunk;

        // ---- async copy of this chunk's 32 KB B panel into LDS ----
        {
#pragma unroll
            for (int ii = 0; ii < 8; ++ii) {
                const int fidx = ii * 1024 + tid * 4;               // float index
                const unsigned dst = b_lds_base + (unsigned)(fidx * 4);
                const unsigned voff = (unsigned)((i0 * kPack + fidx) * 4);
                asm volatile("global_load_async_to_lds_b128 %0, %1, %2"
                             :
                             : "v"(dst), "v"(voff), "s"(bswz)
                             : "memory");
            }
        }

        // ---- stage weighted A chunk (synchronous: needs gamma VALU) ----
        const int s_ = i0 >> 11;                  // residual stream (chunk never crosses)
        const int dbase = i0 & (kD - 1);
        const float* rbase =
            residuals + (((size_t)(b_ * kS + s_) * kN + n_) * kD + dbase);
#pragma unroll
        for (int ii = 0; ii < 4; ++ii) {
            const int kl = ii * 64 + g * 4;       // coalesced across the 16 threads
            const float4 x = *reinterpret_cast<const float4*>(rbase + kl);
            ssq += x.x * x.x + x.y * x.y + x.z * x.z + x.w * x.w;
            const float4 gm = *reinterpret_cast<const float4*>(gamma + i0 + kl);
            float* a = &a_lds[m * kAStride + kl];
            a[0] = x.x * (gm.x + 1.0f);
            a[1] = x.y * (gm.y + 1.0f);
            a[2] = x.z * (gm.z + 1.0f);
            a[3] = x.w * (gm.w + 1.0f);
        }

        asm volatile("s_wait_asynccnt 0x0" ::: "memory");
        __syncthreads();

        // ---- each wave: 8 x (K=4) WMMA steps over its 32-wide K slice ----
#pragma unroll
        for (int kk = 0; kk < 8; ++kk) {
            const int kb = wave * 32 + kk * 4;
            v2f av;
            av.x = a_lds[mrow * kAStride + kb + khalf];
            av.y = a_lds[mrow * kAStride + kb + khalf + 1];
            const float* bp = &b_lds[bbase + kk * 128];
            const v2f b0 = *reinterpret_cast<const v2f*>(bp);        // cols 0-15
            const v2f b1 = *reinterpret_cast<const v2f*>(bp + 32);   // cols 16-31
            // D = A(16x4,f32) * B(4x16,f32) + C(16x16,f32)
            c0 = __builtin_amdgcn_wmma_f32_16x16x4_f32(false, av, false, b0,
                                                       (short)0, c0, false, false);
            c1 = __builtin_amdgcn_wmma_f32_16x16x4_f32(false, av, false, b1,
                                                       (short)0, c1, false, false);
        }
    }

    ssq_lds[m * 16 + g] = ssq;

    // reduce per-wave C tiles (C layout: vgpr vg, lanes 0-15 -> M=vg, 16-31 -> M=8+vg)
    const int Mbase = (lane >= 16) ? 8 : 0;
#pragma unroll
    for (int vg = 0; vg < 8; ++vg) {
        atomicAdd(&c_acc[(vg + Mbase) * 16 + ncol], c0[vg]);
        atomicAdd(&c_acc[256 + (vg + Mbase) * 16 + ncol], c1[vg]);
    }
    __syncthreads();

    // one lane per token finishes: norm scale, alpha/sinkhorn/beta
    if (tid < kTile) {
        const int mm = tid;
        const int tk = tok0 + mm;
        float ss = 0.0f;
#pragma unroll
        for (int q = 0; q < 16; ++q) ss += ssq_lds[mm * 16 + q];
        const float cs = sqrtf((float)kSD) / fmaxf(sqrtf(ss), 1e-12f);

        float dot[kCols];
#pragma unroll
        for (int j = 0; j < 16; ++j) dot[j] = c_acc[mm * 16 + j] * cs;
#pragma unroll
        for (int j = 16; j < kCols; ++j) dot[j] = c_acc[256 + mm * 16 + (j - 16)] * cs;

        const float scl_pre = pre_branch_scale[0];
        const float scl_res = residual_scale[0];
        const float hps = h_post_scale[0];

        float ap[kS];
        float la[kS][kS];
#pragma unroll
        for (int s2 = 0; s2 < kS; ++s2) {
            ap[s2] = sigmoidf_(dot[s2 * 5 + 0] * scl_pre + static_alpha[s2 * 5 + 0]);
#pragma unroll
            for (int t = 1; t <= kS; ++t)
                la[s2][t - 1] = dot[s2 * 5 + t] * scl_res + static_alpha[s2 * 5 + t];
        }
        for (int it = 0; it < kIters; ++it) {
            // column logsumexp (axis=-2)
#pragma unroll
            for (int t = 0; t < kS; ++t) {
                float mx = la[0][t];
                for (int s2 = 1; s2 < kS; ++s2) mx = fmaxf(mx, la[s2][t]);
                float sm = 0.0f;
                for (int s2 = 0; s2 < kS; ++s2) sm += expf(la[s2][t] - mx);
                const float lse = logf(sm) + mx;
                for (int s2 = 0; s2 < kS; ++s2) la[s2][t] -= lse;
            }
            // row logsumexp (axis=-1)
#pragma unroll
            for (int s2 = 0; s2 < kS; ++s2) {
                float mx = la[s2][0];
                for (int t = 1; t < kS; ++t) mx = fmaxf(mx, la[s2][t]);
                float sm = 0.0f;
                for (int t = 0; t < kS; ++t) sm += expf(la[s2][t] - mx);
                const float lse = logf(sm) + mx;
                for (int t = 0; t < kS; ++t) la[s2][t] -= lse;
            }
        }
        float* rec = tokrec + (size_t)tk * kCols;
#pragma unroll
        for (int s2 = 0; s2 < kS; ++s2) {
            rec[s2 * 5 + 0] = ap[s2];
#pragma unroll
            for (int t = 1; t <= kS; ++t) rec[s2 * 5 + t] = expf(la[s2][t - 1]);
            rec[kAlphaCols + s2] =
                2.0f * sigmoidf_(dot[kAlphaCols + s2] * hps + static_beta[s2]);
        }
    }
}

// ---------------------------------------------------------------------------
// Kernel 3: per-token mix + beta combine (streaming, float4, coalesced).
// mix[t,d] = sum_s alpha_full[s,t] * r[s,d]
// out[s,d] = mix[0,d]*beta[s] + mix[s+1,d]
// ---------------------------------------------------------------------------
__global__ __launch_bounds__(256) void mchc_mix(const float* __restrict__ residuals,
                                                const float* __restrict__ tokrec,
                                                float* __restrict__ out) {
    const int token = blockIdx.x;
    const int b_ = token >> 12;
    const int n_ = token & (kN - 1);
    const float* rec = tokrec + (size_t)token * kCols;

    float af[kS][kS + kV];
    float bt[kS];
#pragma unroll
    for (int s2 = 0; s2 < kS; ++s2) {
#pragma unroll
        for (int t = 0; t <= kS; ++t) af[s2][t] = rec[s2 * 5 + t];
        bt[s2] = rec[kAlphaCols + s2];
    }

    const size_t rowstride = (size_t)kN * kD;
    const float* rb = residuals + ((size_t)b_ * kS * kN + n_) * kD;
    float* ob = out + ((size_t)b_ * kS * kN + n_) * kD;

#pragma unroll
    for (int it = 0; it < kD / (256 * 4); ++it) {
        const int d0 = it * 1024 + threadIdx.x * 4;
        float4 r[kS];
#pragma unroll
        for (int s2 = 0; s2 < kS; ++s2)
            r[s2] = *reinterpret_cast<const float4*>(rb + (size_t)s2 * rowstride + d0);

        float4 mix0 = make_float4(0.f, 0.f, 0.f, 0.f);
#pragma unroll
        for (int s2 = 0; s2 < kS; ++s2) mix0 = f4_fma(af[s2][0], r[s2], mix0);

#pragma unroll
        for (int t = 1; t <= kS; ++t) {
            float4 mt = make_float4(0.f, 0.f, 0.f, 0.f);
#pragma unroll
            for (int s2 = 0; s2 < kS; ++s2) mt = f4_fma(af[s2][t], r[s2], mt);
            float4 o = f4_fma(bt[t - 1], mix0, mt);
            *reinterpret_cast<float4*>(ob + (size_t)(t - 1) * rowstride + d0) = o;
        }
    }
}

extern "C" void kernel_launch(void* const* d_in, const int* in_sizes, int n_in,
                              void* d_out, int out_size, void* d_ws, size_t ws_size,
                              hipStream_t stream) {
    const float* residuals       = (const float*)d_in[0];
    const float* gamma           = (const float*)d_in[1];
    const float* dyn_alpha_fn    = (const float*)d_in[2];
    const float* static_alpha    = (const float*)d_in[3];
    const float* pre_branch      = (const float*)d_in[4];
    const float* residual_scale  = (const float*)d_in[5];
    const float* dyn_beta_fn     = (const float*)d_in[6];
    const float* static_beta     = (const float*)d_in[7];
    const float* h_post_scale    = (const float*)d_in[8];
    float* out = (float*)d_out;

    float* bswz   = (float*)d_ws;                        // 8192*32 floats = 1 MB
    float* tokrec = bswz + (size_t)kSD * kPack;          // 8192*24 floats = 768 KB

    mchc_pack<<<(kSD * kPack + 255) / 256, 256, 0, stream>>>(dyn_alpha_fn, dyn_beta_fn,
                                                             bswz);
    mchc_gemv_sinkhorn<<<kTokens / kTile, 256, 0, stream>>>(
        residuals, gamma, bswz, static_alpha, pre_branch, residual_scale,
        static_beta, h_post_scale, tokrec);
    mchc_mix<<<kTokens, 256, 0, stream>>>(residuals, tokrec, out);
}